// FCD_48301202211328
// MI455X (gfx1250) — compile-verified
//
#include <hip/hip_runtime.h>
#include <hip/hip_bf16.h>

typedef __attribute__((ext_vector_type(2))) float v2f;
typedef __attribute__((ext_vector_type(8))) float v8f;

#define EPS   1e-5f
#define SLOPE 0.2f

__device__ __forceinline__ float wave_reduce_sum(float v) {
    // wave32 butterfly reduction
    v += __shfl_xor(v, 16, 32);
    v += __shfl_xor(v,  8, 32);
    v += __shfl_xor(v,  4, 32);
    v += __shfl_xor(v,  2, 32);
    v += __shfl_xor(v,  1, 32);
    return v;
}

// InstanceNorm (biased var) + LeakyReLU(0.2) in-place over LDS rows of width W.
// 8 waves, 16 rows -> wave w handles rows 2w, 2w+1. W in {128,64,32}.
__device__ __forceinline__ void norm_leaky(float* buf, int W, int wave, int lane) {
#pragma unroll
    for (int rr = 0; rr < 2; ++rr) {
        const int m = 2 * wave + rr;
        float s = 0.f, s2 = 0.f;
        for (int j = lane; j < W; j += 32) {
            float v = buf[m * W + j];
            s += v; s2 += v * v;
        }
        s  = wave_reduce_sum(s);
        s2 = wave_reduce_sum(s2);
        const float mean = s / (float)W;
        const float var  = s2 / (float)W - mean * mean;
        const float rinv = rsqrtf(var + EPS);
        for (int j = lane; j < W; j += 32) {
            float v = (buf[m * W + j] - mean) * rinv;
            buf[m * W + j] = (v >= 0.f) ? v : SLOPE * v;
        }
    }
}

// One 16x16 output tile per participating wave: D = A[16xK] * W[ocols x K]^T + bias.
// A lives in LDS (row-major, stride K). W/bias from global (L2-resident, tiny).
// Writes result (+bias) into dst LDS, row-major with stride NWAVES*16.
template <int K, int NWAVES>
__device__ __forceinline__ void fc_tile(const float* A, const float* __restrict__ W,
                                        const float* __restrict__ bias, float* dst,
                                        int wave, int lane) {
    if (wave < NWAVES) {   // wave-uniform: EXEC stays all-ones inside (WMMA requirement)
        const int half = lane >> 4;
        const int l16  = lane & 15;
        v8f acc = {};
        const float* ap = A + l16 * K + 2 * half;
        const float* bp = W + (size_t)(wave * 16 + l16) * K + 2 * half;
#pragma unroll
        for (int k = 0; k < K; k += 4) {
            v2f a = *(const v2f*)(ap + k);
            v2f b = *(const v2f*)(bp + k);
            acc = __builtin_amdgcn_wmma_f32_16x16x4_f32(false, a, false, b,
                                                        (short)0, acc, false, false);
        }
        const float bv = bias[wave * 16 + l16];
#pragma unroll
        for (int r = 0; r < 8; ++r)
            dst[(r + 8 * half) * (NWAVES * 16) + wave * 16 + l16] = acc[r] + bv;
    }
}

__global__ __launch_bounds__(256) void fused_mlp_kernel(
    const float* __restrict__ x1, const float* __restrict__ x2,
    const float* __restrict__ W1, const float* __restrict__ b1,
    const float* __restrict__ W2, const float* __restrict__ b2,
    const float* __restrict__ W3, const float* __restrict__ b3,
    const float* __restrict__ W4, const float* __restrict__ b4,
    float* __restrict__ out) {
    constexpr int D1 = 20000, DIN = 40000, KQ = 10000;

    // 4 K-quarter partials of h1[16][128] = 32 KB. Regions are reused after the
    // combine: hA = combined h1 (later h3), hB = h2.
    __shared__ __align__(16) float pbuf[4 * 2048];
    float* hA = pbuf;          // h1 combined -> later h3 [16][32]
    float* hB = pbuf + 2048;   // h2 [16][64]

    const int tid  = threadIdx.x;
    const int wave = tid >> 5;
    const int lane = tid & 31;
    const int half = lane >> 4;
    const int l16  = lane & 15;
    const int row0 = blockIdx.x << 4;   // 16 rows per workgroup

    // ---------------- Phase 1: h1[16x128] = concat(x1,x2) @ W1^T ----------------
    // Wave split: kw = K-quarter (x1 lo/hi, x2 lo/hi), nw = 64-col group (4 tiles).
    // Per k-step: 1 A-frag load + 4 B-frag loads -> 4 WMMAs (1.25 loads/WMMA).
    const int kw = wave >> 1;           // 0..3
    const int nw = wave & 1;            // output columns [64*nw, 64*nw+64)

    v8f acc0 = {}, acc1 = {}, acc2 = {}, acc3 = {};
    {
        const float* xbase = (kw >= 2) ? x2 : x1;
        const float* ap = xbase + (size_t)(row0 + l16) * D1 + (size_t)(kw & 1) * KQ + 2 * half;
        const float* bp = W1 + (size_t)(nw * 64 + l16) * DIN + (size_t)kw * KQ + 2 * half;
#pragma unroll 4
        for (int k = 0; k < KQ; k += 4) {
            v2f a  = *(const v2f*)ap; ap += 4;
            v2f b0 = *(const v2f*)(bp);
            v2f b1 = *(const v2f*)(bp + (size_t)16 * DIN);
            v2f b2 = *(const v2f*)(bp + (size_t)32 * DIN);
            v2f b3 = *(const v2f*)(bp + (size_t)48 * DIN);
            bp += 4;
            acc0 = __builtin_amdgcn_wmma_f32_16x16x4_f32(false, a, false, b0,
                                                         (short)0, acc0, false, false);
            acc1 = __builtin_amdgcn_wmma_f32_16x16x4_f32(false, a, false, b1,
                                                         (short)0, acc1, false, false);
            acc2 = __builtin_amdgcn_wmma_f32_16x16x4_f32(false, a, false, b2,
                                                         (short)0, acc2, false, false);
            acc3 = __builtin_amdgcn_wmma_f32_16x16x4_f32(false, a, false, b3,
                                                         (short)0, acc3, false, false);
        }
    }
    // Write the 4 K-quarter partials to disjoint LDS regions.
    {
        float* pp = pbuf + kw * 2048;
        const int c0 = nw * 64 + l16;
#pragma unroll
        for (int r = 0; r < 8; ++r) {
            const int ro = (r + 8 * half) * 128;
            pp[ro + c0]      = acc0[r];
            pp[ro + c0 + 16] = acc1[r];
            pp[ro + c0 + 32] = acc2[r];
            pp[ro + c0 + 48] = acc3[r];
        }
    }
    __syncthreads();
    // Combine K-quarters + bias (bank-conflict-free: lane-stride-1).
#pragma unroll
    for (int j = 0; j < 8; ++j) {
        const int i = tid + j * 256;
        float s = pbuf[i] + pbuf[i + 2048] + pbuf[i + 4096] + pbuf[i + 6144];
        pbuf[i] = s + b1[i & 127];
    }
    __syncthreads();
    norm_leaky(hA, 128, wave, lane);
    __syncthreads();

    // ---------------- Phase 2: h2[16x64] = n1 @ W2^T ----------------
    fc_tile<128, 4>(hA, W2, b2, hB, wave, lane);
    __syncthreads();
    norm_leaky(hB, 64, wave, lane);
    __syncthreads();

    // ---------------- Phase 3: h3[16x32] = n2 @ W3^T ----------------
    fc_tile<64, 2>(hB, W3, b3, hA, wave, lane);
    __syncthreads();
    norm_leaky(hA, 32, wave, lane);
    __syncthreads();

    // ---------------- Phase 4: out[16x16] = n3 @ W4^T + b4 ----------------
    if (wave == 0) {
        v8f acc4 = {};
        const float* ap = hA + l16 * 32 + 2 * half;
        const float* bp = W4 + (size_t)l16 * 32 + 2 * half;
#pragma unroll
        for (int k = 0; k < 32; k += 4) {
            v2f a = *(const v2f*)(ap + k);
            v2f b = *(const v2f*)(bp + k);
            acc4 = __builtin_amdgcn_wmma_f32_16x16x4_f32(false, a, false, b,
                                                         (short)0, acc4, false, false);
        }
        const float bv = b4[l16];
#pragma unroll
        for (int r = 0; r < 8; ++r)
            out[(size_t)(row0 + r + 8 * half) * 16 + l16] = acc4[r] + bv;
    }
}

extern "C" void kernel_launch(void* const* d_in, const int* in_sizes, int n_in,
                              void* d_out, int out_size, void* d_ws, size_t ws_size,
                              hipStream_t stream) {
    const float* x1 = (const float*)d_in[0];
    const float* x2 = (const float*)d_in[1];
    const float* W1 = (const float*)d_in[2];
    const float* b1 = (const float*)d_in[3];
    const float* W2 = (const float*)d_in[4];
    const float* b2 = (const float*)d_in[5];
    const float* W3 = (const float*)d_in[6];
    const float* b3 = (const float*)d_in[7];
    const float* W4 = (const float*)d_in[8];
    const float* b4 = (const float*)d_in[9];
    float* out = (float*)d_out;

    // 4096 rows / 16 rows per WG = 256 workgroups, 8 waves each (wave32)
    fused_mlp_kernel<<<256, 256, 0, stream>>>(x1, x2, W1, b1, W2, b2, W3, b3,
                                              W4, b4, out);
    (void)in_sizes; (void)n_in; (void)out_size; (void)d_ws; (void)ws_size;
}